// RNNModel_41317585387881
// MI455X (gfx1250) — compile-verified
//
#include <hip/hip_runtime.h>
#include <math.h>

// Problem constants (from the reference): T=26246, B=128, IN=6, H=1.
#define T_LEN  26246
#define BATCH  128
#define IN_F   6

typedef float v2f __attribute__((ext_vector_type(2)));
typedef float v8f __attribute__((ext_vector_type(8)));

__device__ __forceinline__ float fast_tanh(float v) {
#if __has_builtin(__builtin_amdgcn_tanhf)
    return __builtin_amdgcn_tanhf(v);   // native v_tanh_f32 on gfx1250
#else
    return tanhf(v);
#endif
}

// ---------------------------------------------------------------------------
// Kernel 1: xp[r] = dot(x[r,0:6], Wih) + b_ih + b_hh   for r in [0, T*B)
//
// Implemented with V_WMMA_F32_16X16X4_F32. Mapping (one 16-row tile / wave):
//   A[m][k] = Wih[k]      (broadcast over rows m)          -> 2 VGPRs (v2f)
//   B[k][n] = x[r0+n][k]  (16 x-rows as the 16 columns)    -> 2 VGPRs (v2f)
//   C[m][n] = b_ih + b_hh
// Then D[m][n] = xp[r0+n] for every m. K=6 handled as two chained K=4 WMMAs
// (second one zero-padded in A, so the junk B values multiply by 0).
// ISA VGPR layouts (CDNA5 ISA, 32-bit 16x4 / 16x16):
//   A: lane l -> row m = l%16, VGPR v -> K = v + 2*(l/16)
//   B: lane l -> col n = l%16, VGPR v -> K = v + 2*(l/16)
//   D: lane l<16 holds col n=l in all 8 VGPRs (rows identical here)
// ---------------------------------------------------------------------------
__global__ __launch_bounds__(256) void rnn_xproj_wmma(
    const float* __restrict__ x, const float* __restrict__ Wih,
    const float* __restrict__ bih, const float* __restrict__ bhh,
    float* __restrict__ xp, int nTiles)
{
    const int lane = threadIdx.x & 31;
    const int wave = (int)(blockIdx.x * (blockDim.x >> 5) + (threadIdx.x >> 5));
    const int nWaves = (int)(gridDim.x * (blockDim.x >> 5));

    const int n  = lane & 15;   // column index within the tile
    const int hh = lane >> 4;   // lane half: selects K pairs {2h,2h+1}

    const float bias = bih[0] + bhh[0];

#if __has_builtin(__builtin_amdgcn_wmma_f32_16x16x4_f32)
    // A operands are tile-invariant: hoist them.
    v2f a1, a2;
    a1.x = Wih[2 * hh];
    a1.y = Wih[2 * hh + 1];
    if (hh == 0) { a2.x = Wih[4]; a2.y = Wih[5]; }   // features 4,5
    else         { a2.x = 0.0f;  a2.y = 0.0f;  }     // pad features 6,7

    for (int tile = wave; tile < nTiles; tile += nWaves) {
        const int r = tile * 16 + n;                 // x row this lane feeds
        const float* __restrict__ xr = x + r * IN_F;

        v2f b1, b2;
        b1.x = xr[2 * hh];                           // features {0,1} / {2,3}
        b1.y = xr[2 * hh + 1];
        if (hh == 0) { b2.x = xr[4]; b2.y = xr[5]; } // features {4,5}
        else         { b2.x = 0.0f; b2.y = 0.0f; }   // padded (A2 is 0 here)

        v8f c;
#pragma unroll
        for (int i = 0; i < 8; ++i) c[i] = bias;

        v8f d = __builtin_amdgcn_wmma_f32_16x16x4_f32(
            false, a1, false, b1, (short)0, c, false, false);
        d = __builtin_amdgcn_wmma_f32_16x16x4_f32(
            false, a2, false, b2, (short)0, d, false, false);

        if (lane < 16) xp[tile * 16 + lane] = d[0];
    }
#else
    // Scalar fallback (kept compile-safe if the builtin is ever absent).
    for (int tile = wave; tile < nTiles; tile += nWaves) {
        const int r = tile * 16 + n;
        const float* __restrict__ xr = x + r * IN_F;
        float s = bias;
#pragma unroll
        for (int f = 0; f < IN_F; ++f) s = fmaf(Wih[f], xr[f], s);
        if (lane < 16) xp[tile * 16 + lane] = s;
    }
#endif
}

// ---------------------------------------------------------------------------
// Kernel 2: the sequential recurrence + trailing linear.
// One block of 128 threads (4 wave32s -> one per SIMD of a WGP); thread b owns
// batch chain b. Unroll by 8 with next-chunk prefetch so the global loads sit
// outside the fma->tanh dependency chain. xp is L2-resident (13.4MB << 192MB).
// ---------------------------------------------------------------------------
__global__ __launch_bounds__(128) void rnn_scan_fc(
    const float* __restrict__ xp, const float* __restrict__ Whh,
    const float* __restrict__ fcw, const float* __restrict__ fcb,
    float* __restrict__ out)
{
    const int b = threadIdx.x;           // 0..127
    const float w  = Whh[0];
    const float fw = fcw[0];
    const float fb = fcb[0];

    constexpr int U = 8;
    constexpr int C = T_LEN / U;         // 3280 full chunks, 6-step tail

    float h = 0.0f;
    float buf[U];
#pragma unroll
    for (int u = 0; u < U; ++u) buf[u] = xp[u * BATCH + b];

    for (int c = 0; c < C; ++c) {
        const int base = c * U;
        float nxt[U];
        const bool more = (c + 1) < C;   // wave-uniform
#pragma unroll
        for (int u = 0; u < U; ++u)
            nxt[u] = more ? xp[(base + U + u) * BATCH + b] : 0.0f;

#pragma unroll
        for (int u = 0; u < U; ++u) {
            h = fast_tanh(fmaf(w, h, buf[u]));
            out[(base + u) * BATCH + b] = fmaf(fw, h, fb);
        }

#pragma unroll
        for (int u = 0; u < U; ++u) buf[u] = nxt[u];
    }

    for (int t = C * U; t < T_LEN; ++t) {
        h = fast_tanh(fmaf(w, h, xp[t * BATCH + b]));
        out[t * BATCH + b] = fmaf(fw, h, fb);
    }
}

// ---------------------------------------------------------------------------
// Fallback if the workspace can't hold xp: fuse the 6-wide dot into the scan.
// ---------------------------------------------------------------------------
__global__ __launch_bounds__(128) void rnn_fused(
    const float* __restrict__ x, const float* __restrict__ Wih,
    const float* __restrict__ bih, const float* __restrict__ bhh,
    const float* __restrict__ Whh, const float* __restrict__ fcw,
    const float* __restrict__ fcb, float* __restrict__ out)
{
    const int b = threadIdx.x;
    const float bias = bih[0] + bhh[0];
    float w0 = Wih[0], w1 = Wih[1], w2 = Wih[2],
          w3 = Wih[3], w4 = Wih[4], w5 = Wih[5];
    const float w  = Whh[0];
    const float fw = fcw[0];
    const float fb = fcb[0];

    float h = 0.0f;
    for (int t = 0; t < T_LEN; ++t) {
        const float* __restrict__ xr = x + (t * BATCH + b) * IN_F;
        float a = bias;
        a = fmaf(w0, xr[0], a); a = fmaf(w1, xr[1], a);
        a = fmaf(w2, xr[2], a); a = fmaf(w3, xr[3], a);
        a = fmaf(w4, xr[4], a); a = fmaf(w5, xr[5], a);
        h = fast_tanh(fmaf(w, h, a));
        out[t * BATCH + b] = fmaf(fw, h, fb);
    }
}

extern "C" void kernel_launch(void* const* d_in, const int* in_sizes, int n_in,
                              void* d_out, int out_size, void* d_ws, size_t ws_size,
                              hipStream_t stream)
{
    (void)in_sizes; (void)n_in; (void)out_size;

    const float* x   = (const float*)d_in[0];
    const float* Wih = (const float*)d_in[1];
    const float* Whh = (const float*)d_in[2];
    const float* bih = (const float*)d_in[3];
    const float* bhh = (const float*)d_in[4];
    const float* fcw = (const float*)d_in[5];
    const float* fcb = (const float*)d_in[6];
    float* out = (float*)d_out;

    const size_t need = (size_t)T_LEN * BATCH * sizeof(float);  // 13.4 MB
    if (ws_size >= need) {
        float* xp = (float*)d_ws;
        const int nTiles = (T_LEN * BATCH) / 16;                // 209968 exact
        rnn_xproj_wmma<<<1024, 256, 0, stream>>>(x, Wih, bih, bhh, xp, nTiles);
        rnn_scan_fc<<<1, 128, 0, stream>>>(xp, Whh, fcw, fcb, out);
    } else {
        rnn_fused<<<1, 128, 0, stream>>>(x, Wih, bih, bhh, Whh, fcw, fcb, out);
    }
}